// GAT_11141145166398
// MI455X (gfx1250) — compile-verified
//
#include <hip/hip_runtime.h>
#include <hip/hip_bf16.h>

typedef __attribute__((ext_vector_type(2))) float v2f;
typedef __attribute__((ext_vector_type(8))) float v8f;

#define IN_CH 128
#define HC    256
#define NHEAD 4
#define CH    64
#define NEG_SLOPE 0.2f
#define BN_EPS    1e-5f

// ---- helpers -------------------------------------------------------------

// monotone float<->uint map so atomicMax(u32) == float max
__device__ __forceinline__ unsigned enc_f32(float f) {
    unsigned u = __float_as_uint(f);
    return (u & 0x80000000u) ? ~u : (u | 0x80000000u);
}
__device__ __forceinline__ float dec_f32(unsigned u) {
    return (u & 0x80000000u) ? __uint_as_float(u & 0x7FFFFFFFu)
                             : __uint_as_float(~u);
}
#define ENC_NEG_INF 0x007FFFFFu   // enc_f32(-inf)

__device__ __forceinline__ float lrelu(float x) { return x > 0.f ? x : NEG_SLOPE * x; }

// ---- generic fill --------------------------------------------------------
__global__ void fill_u32_kernel(unsigned* __restrict__ p, unsigned v, long long count) {
    long long i = (long long)blockIdx.x * blockDim.x + threadIdx.x;
    long long stride = (long long)gridDim.x * blockDim.x;
    for (; i < count; i += stride) p[i] = v;
}

// ---- layer-1 GEMM: h1[N,256] = x[N,128] @ W1[128,256], f32 WMMA ----------
// block = 128 threads (4 waves); blockIdx.x = M-tile (16 rows);
// wave w computes columns [w*64, w*64+64).
__global__ void gemm1_kernel(const float* __restrict__ x,
                             const float* __restrict__ W,
                             float* __restrict__ h, int n) {
    int mt   = blockIdx.x;
    int wave = threadIdx.x >> 5;
    int lane = threadIdx.x & 31;
    int m0 = mt * 16;
    int n0 = wave * 64;
    int row = lane & 15;          // A-row (M) and B/C column (N) sub-index
    int kg  = (lane >> 4) << 1;   // 0 or 2

    int mrow = m0 + row; if (mrow >= n) mrow = n - 1;   // clamp, keep EXEC uniform
    const float* xr = x + (long long)mrow * IN_CH;

    v8f acc0 = {}, acc1 = {}, acc2 = {}, acc3 = {};
    #pragma unroll 4
    for (int k0 = 0; k0 < IN_CH; k0 += 4) {
        int k = k0 + kg;
        v2f a; a.x = xr[k]; a.y = xr[k + 1];
        const float* w0 = W + (long long)k * HC + n0 + row;
        const float* w1 = W + (long long)(k + 1) * HC + n0 + row;
        v2f b0; b0.x = w0[0];  b0.y = w1[0];
        v2f b1; b1.x = w0[16]; b1.y = w1[16];
        v2f b2; b2.x = w0[32]; b2.y = w1[32];
        v2f b3; b3.x = w0[48]; b3.y = w1[48];
        acc0 = __builtin_amdgcn_wmma_f32_16x16x4_f32(false, a, false, b0, (short)0, acc0, false, false);
        acc1 = __builtin_amdgcn_wmma_f32_16x16x4_f32(false, a, false, b1, (short)0, acc1, false, false);
        acc2 = __builtin_amdgcn_wmma_f32_16x16x4_f32(false, a, false, b2, (short)0, acc2, false, false);
        acc3 = __builtin_amdgcn_wmma_f32_16x16x4_f32(false, a, false, b3, (short)0, acc3, false, false);
    }
    int r0  = m0 + ((lane >> 4) << 3);   // C/D: M = v + 8*(lane>>4)
    int col = n0 + row;                  // C/D: N = lane&15
    #pragma unroll
    for (int v = 0; v < 8; v++) {
        int r = r0 + v;
        if (r < n) {
            float* hr = h + (long long)r * HC + col;
            hr[0]  = acc0[v];
            hr[16] = acc1[v];
            hr[32] = acc2[v];
            hr[48] = acc3[v];
        }
    }
}

// ---- per-(node,head) attention coefficients ------------------------------
__global__ void alpha1_kernel(const float* __restrict__ h1,
                              const float* __restrict__ a_s,
                              const float* __restrict__ a_d,
                              float* __restrict__ as1, float* __restrict__ ad1, int n) {
    int t = blockIdx.x * blockDim.x + threadIdx.x;
    if (t >= n * NHEAD) return;
    int node = t >> 2, hd = t & 3;
    const float* hp  = h1 + (long long)node * HC + hd * CH;
    const float* asw = a_s + hd * CH;
    const float* adw = a_d + hd * CH;
    float sa = 0.f, sd = 0.f;
    #pragma unroll 8
    for (int c = 0; c < CH; c++) { float v = hp[c]; sa += v * asw[c]; sd += v * adw[c]; }
    as1[node * 4 + hd] = sa;
    ad1[node * 4 + hd] = sd;
}

// ---- layer-1 edge pass A: segment max ------------------------------------
__global__ void edgeA1_kernel(const int* __restrict__ ei,
                              const float4* __restrict__ AS, const float4* __restrict__ AD,
                              unsigned* __restrict__ amax, int E, int total) {
    int e = blockIdx.x * blockDim.x + threadIdx.x;
    if (e >= total) return;
    int src, dst;
    if (e < E) { src = ei[e]; dst = ei[E + e]; } else { src = dst = e - E; }
    float4 a = AS[src], d = AD[dst];
    unsigned* m = amax + (long long)dst * 4;
    atomicMax(&m[0], enc_f32(lrelu(a.x + d.x)));
    atomicMax(&m[1], enc_f32(lrelu(a.y + d.y)));
    atomicMax(&m[2], enc_f32(lrelu(a.z + d.z)));
    atomicMax(&m[3], enc_f32(lrelu(a.w + d.w)));
}

// ---- layer-1 edge pass B: exp-sum + unnormalized aggregation -------------
// one wave (32 lanes) per edge; 8 channels per lane, head = j>>1
__global__ void edgeB1_kernel(const int* __restrict__ ei,
                              const float4* __restrict__ AS, const float4* __restrict__ AD,
                              const uint4* __restrict__ amax,
                              const float* __restrict__ h1,
                              float* __restrict__ den, float* __restrict__ num,
                              int E, int total) {
    int gw   = (blockIdx.x * blockDim.x + threadIdx.x) >> 5;
    int lane = threadIdx.x & 31;
    if (gw >= total) return;
    int src, dst;
    if (gw < E) { src = ei[gw]; dst = ei[E + gw]; } else { src = dst = gw - E; }
    float4 a = AS[src], d = AD[dst];
    uint4  m = amax[dst];
    float ex0 = expf(lrelu(a.x + d.x) - dec_f32(m.x));
    float ex1 = expf(lrelu(a.y + d.y) - dec_f32(m.y));
    float ex2 = expf(lrelu(a.z + d.z) - dec_f32(m.z));
    float ex3 = expf(lrelu(a.w + d.w) - dec_f32(m.w));
    if (lane == 0) {
        float* dn = den + (long long)dst * 4;
        atomicAdd(&dn[0], ex0); atomicAdd(&dn[1], ex1);
        atomicAdd(&dn[2], ex2); atomicAdd(&dn[3], ex3);
    }
    const float* hs = h1 + (long long)src * HC;
    float*       nd = num + (long long)dst * HC;
    atomicAdd(&nd[lane       ], hs[lane       ] * ex0);
    atomicAdd(&nd[lane +  32 ], hs[lane +  32 ] * ex0);
    atomicAdd(&nd[lane +  64 ], hs[lane +  64 ] * ex1);
    atomicAdd(&nd[lane +  96 ], hs[lane +  96 ] * ex1);
    atomicAdd(&nd[lane + 128 ], hs[lane + 128 ] * ex2);
    atomicAdd(&nd[lane + 160 ], hs[lane + 160 ] * ex2);
    atomicAdd(&nd[lane + 192 ], hs[lane + 192 ] * ex3);
    atomicAdd(&nd[lane + 224 ], hs[lane + 224 ] * ex3);
}

// ---- normalize + bias + BN(eval) + ELU, in place (num1 -> h2) ------------
__global__ void finish1_kernel(float* __restrict__ num, const float* __restrict__ den,
                               const float* __restrict__ b1,
                               const float* __restrict__ gamma, const float* __restrict__ beta,
                               long long total) {
    long long i = (long long)blockIdx.x * blockDim.x + threadIdx.x;
    if (i >= total) return;
    int c = (int)(i & (HC - 1));
    long long node = i >> 8;
    float v = num[i] / den[node * 4 + (c >> 6)] + b1[c];
    v = v * (gamma[c] * rsqrtf(1.f + BN_EPS)) + beta[c];
    v = v > 0.f ? v : (expf(v) - 1.f);
    num[i] = v;
}

// ---- layer-2 matvec: z[n] = h2[n,:] . W2[:,0]  (wave per node) -----------
__global__ void node2_kernel(const float* __restrict__ h2, const float* __restrict__ W2,
                             float* __restrict__ z, int n) {
    int gw   = (blockIdx.x * blockDim.x + threadIdx.x) >> 5;
    int lane = threadIdx.x & 31;
    if (gw >= n) return;
    const float* hp = h2 + (long long)gw * HC;
    float s = 0.f;
    #pragma unroll
    for (int j = 0; j < 8; j++) s += hp[lane + 32 * j] * W2[lane + 32 * j];
    #pragma unroll
    for (int off = 16; off > 0; off >>= 1) s += __shfl_xor(s, off, 32);
    if (lane == 0) z[gw] = s;
}

// ---- layer-2 edge passes (scalar head) -----------------------------------
__global__ void edgeA2_kernel(const int* __restrict__ ei, const float* __restrict__ z,
                              const float* __restrict__ asc, const float* __restrict__ adc,
                              unsigned* __restrict__ amax, int E, int total) {
    int e = blockIdx.x * blockDim.x + threadIdx.x;
    if (e >= total) return;
    int src, dst;
    if (e < E) { src = ei[e]; dst = ei[E + e]; } else { src = dst = e - E; }
    float al = lrelu(z[src] * asc[0] + z[dst] * adc[0]);
    atomicMax(&amax[dst], enc_f32(al));
}

__global__ void edgeB2_kernel(const int* __restrict__ ei, const float* __restrict__ z,
                              const float* __restrict__ asc, const float* __restrict__ adc,
                              const unsigned* __restrict__ amax,
                              float* __restrict__ den, float* __restrict__ num,
                              int E, int total) {
    int e = blockIdx.x * blockDim.x + threadIdx.x;
    if (e >= total) return;
    int src, dst;
    if (e < E) { src = ei[e]; dst = ei[E + e]; } else { src = dst = e - E; }
    float al = lrelu(z[src] * asc[0] + z[dst] * adc[0]);
    float ex = expf(al - dec_f32(amax[dst]));
    atomicAdd(&den[dst], ex);
    atomicAdd(&num[dst], ex * z[src]);
}

// ---- final: bias + MLP(1->64->1) + sigmoid -------------------------------
__global__ void final_kernel(const float* __restrict__ num, const float* __restrict__ den,
                             const float* __restrict__ b2,
                             const float* __restrict__ w1, const float* __restrict__ mb1,
                             const float* __restrict__ w2, const float* __restrict__ mb2,
                             float* __restrict__ out, int n) {
    int t = blockIdx.x * blockDim.x + threadIdx.x;
    if (t >= n) return;
    float o = num[t] / den[t] + b2[0];
    float acc = mb2[0];
    #pragma unroll 8
    for (int c = 0; c < CH; c++) {
        float r = o * w1[c] + mb1[c];
        r = r > 0.f ? r : 0.f;
        acc += r * w2[c];
    }
    out[t] = 1.f / (1.f + expf(-acc));
}

// ---- host launcher -------------------------------------------------------
extern "C" void kernel_launch(void* const* d_in, const int* in_sizes, int n_in,
                              void* d_out, int out_size, void* d_ws, size_t ws_size,
                              hipStream_t stream) {
    const float* x        = (const float*)d_in[0];
    const int*   ei       = (const int*)  d_in[1];
    const float* W1       = (const float*)d_in[2];
    const float* a_src1   = (const float*)d_in[3];
    const float* a_dst1   = (const float*)d_in[4];
    const float* b1       = (const float*)d_in[5];
    const float* bn_gamma = (const float*)d_in[6];
    const float* bn_beta  = (const float*)d_in[7];
    const float* W2       = (const float*)d_in[8];
    const float* a_src2   = (const float*)d_in[9];
    const float* a_dst2   = (const float*)d_in[10];
    const float* b2       = (const float*)d_in[11];
    const float* mlp_w1   = (const float*)d_in[12];
    const float* mlp_b1   = (const float*)d_in[13];
    const float* mlp_w2   = (const float*)d_in[14];
    const float* mlp_b2   = (const float*)d_in[15];

    int n = in_sizes[0] / IN_CH;     // 100000
    int E = in_sizes[1] / 2;         // 1600000
    int total_e = E + n;

    // workspace bump allocator (256B aligned)
    char* wsb = (char*)d_ws;
    size_t off = 0;
    auto alloc = [&](size_t bytes) -> void* {
        void* p = wsb + off;
        off += (bytes + 255) & ~(size_t)255;
        return p;
    };
    float*    h1    = (float*)   alloc((size_t)n * HC * 4);  // 102.4 MB
    float*    num1  = (float*)   alloc((size_t)n * HC * 4);  // 102.4 MB (becomes h2)
    float*    as1   = (float*)   alloc((size_t)n * 4 * 4);
    float*    ad1   = (float*)   alloc((size_t)n * 4 * 4);
    float*    den1  = (float*)   alloc((size_t)n * 4 * 4);
    unsigned* amax1 = (unsigned*)alloc((size_t)n * 4 * 4);
    float*    z     = (float*)   alloc((size_t)n * 4);
    float*    num2  = (float*)   alloc((size_t)n * 4);
    float*    den2  = (float*)   alloc((size_t)n * 4);
    unsigned* amax2 = (unsigned*)alloc((size_t)n * 4);
    (void)ws_size; (void)n_in; (void)out_size;

    // init accumulators
    {
        long long c1 = (long long)n * HC;
        fill_u32_kernel<<<(unsigned)((c1 + 255) / 256), 256, 0, stream>>>((unsigned*)num1, 0u, c1);
        fill_u32_kernel<<<(unsigned)((n * 4 + 255) / 256), 256, 0, stream>>>((unsigned*)den1, 0u, (long long)n * 4);
        fill_u32_kernel<<<(unsigned)((n * 4 + 255) / 256), 256, 0, stream>>>(amax1, ENC_NEG_INF, (long long)n * 4);
        fill_u32_kernel<<<(unsigned)((n + 255) / 256), 256, 0, stream>>>((unsigned*)num2, 0u, n);
        fill_u32_kernel<<<(unsigned)((n + 255) / 256), 256, 0, stream>>>((unsigned*)den2, 0u, n);
        fill_u32_kernel<<<(unsigned)((n + 255) / 256), 256, 0, stream>>>(amax2, ENC_NEG_INF, n);
    }

    int mtiles = (n + 15) / 16;
    gemm1_kernel<<<mtiles, 128, 0, stream>>>(x, W1, h1, n);
    alpha1_kernel<<<(n * NHEAD + 255) / 256, 256, 0, stream>>>(h1, a_src1, a_dst1, as1, ad1, n);
    edgeA1_kernel<<<(total_e + 255) / 256, 256, 0, stream>>>(ei, (const float4*)as1, (const float4*)ad1,
                                                             amax1, E, total_e);
    edgeB1_kernel<<<(total_e + 7) / 8, 256, 0, stream>>>(ei, (const float4*)as1, (const float4*)ad1,
                                                         (const uint4*)amax1, h1, den1, num1, E, total_e);
    {
        long long total = (long long)n * HC;
        finish1_kernel<<<(unsigned)((total + 255) / 256), 256, 0, stream>>>(num1, den1, b1, bn_gamma, bn_beta, total);
    }
    node2_kernel<<<(n + 7) / 8, 256, 0, stream>>>(num1, W2, z, n);
    edgeA2_kernel<<<(total_e + 255) / 256, 256, 0, stream>>>(ei, z, a_src2, a_dst2, amax2, E, total_e);
    edgeB2_kernel<<<(total_e + 255) / 256, 256, 0, stream>>>(ei, z, a_src2, a_dst2, amax2, den2, num2, E, total_e);
    final_kernel<<<(n + 255) / 256, 256, 0, stream>>>(num2, den2, b2, mlp_w1, mlp_b1, mlp_w2, mlp_b2,
                                                      (float*)d_out, n);
}